// MinGRUFused_3427383902208
// MI455X (gfx1250) — compile-verified
//
#include <hip/hip_runtime.h>
#include <hip/hip_bf16.h>
#include <math.h>

// Problem constants from the reference
#define B_   32
#define T_   4096
#define DIN_ 256
#define H_   256
#define NC_  32            // scan chunks along T
#define L_   (T_ / NC_)    // 128 steps per chunk

typedef float v2f __attribute__((ext_vector_type(2)));
typedef float v8f __attribute__((ext_vector_type(8)));

__device__ __forceinline__ float softplus_f(float u) {
    // stable softplus: max(u,0) + log1p(exp(-|u|))
    return fmaxf(u, 0.0f) + log1pf(__expf(-fabsf(u)));
}

__device__ __forceinline__ float logaddexp_f(float a, float b) {
    float mx = fmaxf(a, b);
    float d  = fabsf(a - b);
    return mx + log1pf(__expf(-d));
}

__device__ __forceinline__ float log_g_f(float a) {
    // log g(a): a>=0 -> log(a+0.5), else -softplus(-a)
    return (a >= 0.0f) ? __logf(a + 0.5f) : -softplus_f(-a);
}

// -----------------------------------------------------------------------------
// Kernel 1: fused dual GEMM (k = x Wz^T + bz ; hp = x Wh^T + bh) in exact fp32
// via V_WMMA_F32_16X16X4_F32, plus the log-space elementwise epilogue.
//   v = log_z + log_g(hp) -> d_out ;  c = log_coeffs -> ws
// One wave (32 lanes) computes one 16x16 output tile over full K=256.
// -----------------------------------------------------------------------------
__global__ __launch_bounds__(32) void mingru_gemm_kernel(
    const float* __restrict__ x,  const float* __restrict__ Wz,
    const float* __restrict__ bz, const float* __restrict__ Wh,
    const float* __restrict__ bh, float* __restrict__ v_out,
    float* __restrict__ c_out)
{
    const int lane  = threadIdx.x;         // 0..31, full wave => EXEC all ones
    const int m0    = blockIdx.y << 4;     // row tile over B*T
    const int n0    = blockIdx.x << 4;     // col tile over H
    const int l15   = lane & 15;
    const int khalf = (lane >> 4) << 1;    // K sub-offset 0 or 2

    // fp32 WMMA A-fragment: a[i] = A[M=lane&15][K = khalf+i]
    // fp32 WMMA B-fragment: b[i] = B[K = khalf+i][N = lane&15] = W[n][k]
    const float* xp  = x  + (size_t)(m0 + l15) * DIN_ + khalf;
    const float* wzp = Wz + (size_t)(n0 + l15) * DIN_ + khalf;
    const float* whp = Wh + (size_t)(n0 + l15) * DIN_ + khalf;

    v8f accz = {};   // k-gate accumulator (16x16 f32 C/D layout)
    v8f acch = {};   // h-candidate accumulator

#pragma unroll 4
    for (int kk = 0; kk < DIN_; kk += 4) {
        v2f a  = *(const v2f*)(xp  + kk);
        v2f bz2 = *(const v2f*)(wzp + kk);
        v2f bh2 = *(const v2f*)(whp + kk);
        accz = __builtin_amdgcn_wmma_f32_16x16x4_f32(
                   false, a, false, bz2, (short)0, accz, false, false);
        acch = __builtin_amdgcn_wmma_f32_16x16x4_f32(
                   false, a, false, bh2, (short)0, acch, false, false);
    }

    // Epilogue: C/D layout -> row = m0 + j + (lane>>4)*8, col = n0 + (lane&15)
    const int   ncol = n0 + l15;
    const float bzb  = bz[ncol];
    const float bhb  = bh[ncol];
    const int   mbase = m0 + ((lane >> 4) << 3);

#pragma unroll
    for (int j = 0; j < 8; ++j) {
        float kpre  = accz[j] + bzb;
        float sp    = softplus_f(kpre);
        float log_z = kpre - sp;          // -softplus(-k)
        float log_c = -sp;                // log(1 - sigmoid(k))
        float hp    = acch[j] + bhb;
        float v     = log_z + log_g_f(hp);

        size_t idx = (size_t)(mbase + j) * H_ + ncol;
        v_out[idx] = v;
        c_out[idx] = log_c;
    }
}

// -----------------------------------------------------------------------------
// Kernel 2: per-chunk associative reduction of (c, v) pairs.
//   (C,V) <- (C + c_t, logaddexp(V + c_t, v_t)),  start (0, -inf)
// Grid: (NC_, B_), block: H_ threads (h fastest -> coalesced 1KB lines).
// -----------------------------------------------------------------------------
__global__ __launch_bounds__(256) void mingru_chunk_reduce(
    const float* __restrict__ c_arr, const float* __restrict__ v_arr,
    float* __restrict__ Csum, float* __restrict__ Vsum)
{
    const int h = threadIdx.x;
    const int j = blockIdx.x;   // chunk
    const int b = blockIdx.y;   // batch

    size_t base = ((size_t)b * T_ + (size_t)j * L_) * H_ + h;
    float C = 0.0f;
    float V = -INFINITY;

    for (int t = 0; t < L_; ++t) {
        size_t idx = base + (size_t)t * H_;
        __builtin_prefetch(c_arr + idx + (size_t)8 * H_, 0, 0);
        __builtin_prefetch(v_arr + idx + (size_t)8 * H_, 0, 0);
        float c = c_arr[idx];
        float v = v_arr[idx];
        C += c;
        V = logaddexp_f(V + c, v);
    }
    size_t sidx = ((size_t)b * NC_ + j) * H_ + h;
    Csum[sidx] = C;
    Vsum[sidx] = V;
}

// -----------------------------------------------------------------------------
// Kernel 3: sequential carry scan across the NC_ chunk summaries per channel.
// entry[b][j][h] = scan state at the start of chunk j.
// -----------------------------------------------------------------------------
__global__ __launch_bounds__(256) void mingru_carry(
    const float* __restrict__ h0, const float* __restrict__ Csum,
    const float* __restrict__ Vsum, float* __restrict__ entry)
{
    const int h = threadIdx.x;
    const int b = blockIdx.x;

    float state = log_g_f(h0[(size_t)b * H_ + h]);
#pragma unroll
    for (int j = 0; j < NC_; ++j) {
        size_t sidx = ((size_t)b * NC_ + j) * H_ + h;
        entry[sidx] = state;
        state = logaddexp_f(state + Csum[sidx], Vsum[sidx]);
    }
}

// -----------------------------------------------------------------------------
// Kernel 4: re-scan each chunk from its entry state, write exp(state).
// v lives in d_out and is overwritten in place (read-then-write per element,
// single owner thread -> safe).
// -----------------------------------------------------------------------------
__global__ __launch_bounds__(256) void mingru_apply(
    const float* __restrict__ c_arr, float* __restrict__ vout,
    const float* __restrict__ entry)
{
    const int h = threadIdx.x;
    const int j = blockIdx.x;
    const int b = blockIdx.y;

    float state = entry[((size_t)b * NC_ + j) * H_ + h];
    size_t base = ((size_t)b * T_ + (size_t)j * L_) * H_ + h;

    for (int t = 0; t < L_; ++t) {
        size_t idx = base + (size_t)t * H_;
        __builtin_prefetch(c_arr + idx + (size_t)8 * H_, 0, 0);
        float c = c_arr[idx];
        float v = vout[idx];
        state = logaddexp_f(state + c, v);
        vout[idx] = __expf(state);
    }
}

// -----------------------------------------------------------------------------
// Launch: inputs are (x, h0, Wz, bz, Wh, bh), all fp32. Output fp32 (B,T,H).
// Workspace layout (floats): c[B*T*H] | Csum[B*NC*H] | Vsum[B*NC*H] | entry[B*NC*H]
// -----------------------------------------------------------------------------
extern "C" void kernel_launch(void* const* d_in, const int* in_sizes, int n_in,
                              void* d_out, int out_size, void* d_ws, size_t ws_size,
                              hipStream_t stream)
{
    const float* x  = (const float*)d_in[0];
    const float* h0 = (const float*)d_in[1];
    const float* Wz = (const float*)d_in[2];
    const float* bz = (const float*)d_in[3];
    const float* Wh = (const float*)d_in[4];
    const float* bh = (const float*)d_in[5];
    float* out = (float*)d_out;

    float* ws    = (float*)d_ws;
    float* c_arr = ws;                                     // B*T*H
    float* Csum  = c_arr + (size_t)B_ * T_ * H_;           // B*NC*H
    float* Vsum  = Csum  + (size_t)B_ * NC_ * H_;          // B*NC*H
    float* entry = Vsum  + (size_t)B_ * NC_ * H_;          // B*NC*H

    // Dual GEMM + log-space epilogue: one wave per 16x16 tile
    dim3 ggrid(H_ / 16, (B_ * T_) / 16);                   // (16, 8192)
    mingru_gemm_kernel<<<ggrid, 32, 0, stream>>>(x, Wz, bz, Wh, bh, out, c_arr);

    // Chunked associative scan
    dim3 sgrid(NC_, B_);                                   // (32, 32)
    mingru_chunk_reduce<<<sgrid, H_, 0, stream>>>(c_arr, out, Csum, Vsum);
    mingru_carry<<<B_, H_, 0, stream>>>(h0, Csum, Vsum, entry);
    mingru_apply<<<sgrid, H_, 0, stream>>>(c_arr, out, entry);
}